// SimplePillarEncoder_13288628814455
// MI455X (gfx1250) — compile-verified
//
#include <hip/hip_runtime.h>

typedef __attribute__((ext_vector_type(2))) float v2f;
typedef __attribute__((ext_vector_type(8))) float v8f;

#define P_PILLARS 40000
#define NMAXPT    32
#define IN_CH     9
#define K_PAD     12
#define OUT_CH    64
#define X_L       432
#define Y_L       496
#define VOX       0.16f
#define X_OFF     0.08f          /* VX/2 + 0.0   */
#define Y_OFF     (-39.6f)       /* VY/2 - 39.68 */
#define BN_EPS    1e-3f
#define NSTAT     54             /* 9 first moments + 45 upper-tri second moments */
#define STATS_BLOCKS 250

__device__ __forceinline__ float wave_sum(float v) {
    #pragma unroll
    for (int m = 1; m < 32; m <<= 1) v += __shfl_xor(v, m, 32);
    return v;
}

// Compute the 9 masked features of point `lane` of pillar `p` (wave-cooperative:
// needs all 32 lanes for the center reduction).
__device__ __forceinline__ void pillar_features(
    const float* __restrict__ pillars, const int* __restrict__ coors,
    const int* __restrict__ npoints, int p, int lane, float f[IN_CH])
{
    const float4 pt = ((const float4*)pillars)[p * NMAXPT + lane];
    // center: sum over ALL 32 stored points / npts (matches reference exactly)
    const float sx = wave_sum(pt.x);
    const float sy = wave_sum(pt.y);
    const float sz = wave_sum(pt.z);
    const int   np = npoints[p];
    const float inv_n = 1.0f / (float)np;
    const float cxv = (float)coors[p * 3 + 1] * VOX + X_OFF;
    const float cyv = (float)coors[p * 3 + 2] * VOX + Y_OFF;
    const float mask = (lane < np) ? 1.0f : 0.0f;
    f[0] = pt.x * mask;
    f[1] = pt.y * mask;
    f[2] = pt.z * mask;
    f[3] = pt.w * mask;
    f[4] = (pt.x - sx * inv_n) * mask;
    f[5] = (pt.y - sy * inv_n) * mask;
    f[6] = (pt.z - sz * inv_n) * mask;
    f[7] = (pt.x - cxv) * mask;
    f[8] = (pt.y - cyv) * mask;
}

// ---------------- Pass A: feature moments (9 sums + 45 upper-tri products) ----
__global__ __launch_bounds__(256) void pillar_stats(
    const float* __restrict__ pillars, const int* __restrict__ coors,
    const int* __restrict__ npoints, float* __restrict__ partials)
{
    const int lane = threadIdx.x & 31;
    const int wv   = threadIdx.x >> 5;
    const int wave = blockIdx.x * 8 + wv;
    const int nwaves = STATS_BLOCKS * 8;            // 2000 waves, 20 pillars each

    float acc[NSTAT];
    #pragma unroll
    for (int i = 0; i < NSTAT; i++) acc[i] = 0.0f;

    for (int p = wave; p < P_PILLARS; p += nwaves) {
        float f[IN_CH];
        pillar_features(pillars, coors, npoints, p, lane, f);
        int idx = IN_CH;
        #pragma unroll
        for (int i = 0; i < IN_CH; i++) {
            acc[i] += f[i];
            #pragma unroll
            for (int j = i; j < IN_CH; j++) acc[idx++] += f[i] * f[j];
        }
    }
    // deterministic fixed-tree wave reduction
    #pragma unroll
    for (int i = 0; i < NSTAT; i++) acc[i] = wave_sum(acc[i]);

    __shared__ float sred[8][NSTAT];
    if (lane == 0) {
        #pragma unroll
        for (int i = 0; i < NSTAT; i++) sred[wv][i] = acc[i];
    }
    __syncthreads();
    if (threadIdx.x < NSTAT) {                       // fixed-order block combine
        float s = 0.0f;
        #pragma unroll
        for (int w = 0; w < 8; w++) s += sred[w][threadIdx.x];
        partials[blockIdx.x * NSTAT + threadIdx.x] = s;
    }
}

// ---------------- Finalize: fold BN stats into per-channel scale/shift --------
__global__ __launch_bounds__(64) void pillar_finalize(
    const float* __restrict__ partials, const float* __restrict__ W,
    const float* __restrict__ gamma, const float* __restrict__ beta,
    float* __restrict__ scale, float* __restrict__ shift)
{
    __shared__ float S[NSTAT];
    if (threadIdx.x < NSTAT) {
        float s = 0.0f;
        for (int b = 0; b < STATS_BLOCKS; b++) s += partials[b * NSTAT + threadIdx.x];
        S[threadIdx.x] = s * (1.0f / ((float)P_PILLARS * (float)NMAXPT));
    }
    __syncthreads();
    const int o = threadIdx.x;                       // 0..63
    float w[IN_CH];
    #pragma unroll
    for (int k = 0; k < IN_CH; k++) w[k] = W[o * IN_CH + k];
    float mean = 0.0f;
    #pragma unroll
    for (int k = 0; k < IN_CH; k++) mean += w[k] * S[k];
    float exx = 0.0f;
    #pragma unroll
    for (int i = 0; i < IN_CH; i++) {
        #pragma unroll
        for (int j = 0; j < IN_CH; j++) {
            const int a = (i < j) ? i : j;
            const int b = (i < j) ? j : i;
            exx += w[i] * w[j] * S[IN_CH + IN_CH * a - (a * (a - 1)) / 2 + (b - a)];
        }
    }
    const float var = exx - mean * mean;
    const float sc  = gamma[o] * rsqrtf(var + BN_EPS);
    scale[o] = sc;
    shift[o] = beta[o] - mean * sc;
}

// ---------------- Pass B: WMMA GEMM + BN + ReLU + maxpool + scatter -----------
__global__ __launch_bounds__(256) void pillar_main(
    const float* __restrict__ pillars, const int* __restrict__ coors,
    const int* __restrict__ npoints, const float* __restrict__ W,
    const float* __restrict__ scale, const float* __restrict__ shift,
    float* __restrict__ out)
{
    __shared__ float sW[K_PAD][OUT_CH];              // W transposed, K padded to 12
    __shared__ float sScale[OUT_CH], sShift[OUT_CH];
    __shared__ float sFeat[8][NMAXPT][K_PAD];        // per-wave feature tile

    for (int i = threadIdx.x; i < K_PAD * OUT_CH; i += 256) {
        const int k = i / OUT_CH, o = i % OUT_CH;
        sW[k][o] = (k < IN_CH) ? W[o * IN_CH + k] : 0.0f;
    }
    if (threadIdx.x < OUT_CH) {
        sScale[threadIdx.x] = scale[threadIdx.x];
        sShift[threadIdx.x] = shift[threadIdx.x];
    }

    const int lane = threadIdx.x & 31;
    const int wv   = threadIdx.x >> 5;
    const int p    = blockIdx.x * 8 + wv;            // one pillar per wave (exact)

    float f[IN_CH];
    pillar_features(pillars, coors, npoints, p, lane, f);
    #pragma unroll
    for (int k = 0; k < IN_CH; k++) sFeat[wv][lane][k] = f[k];
    sFeat[wv][lane][9] = 0.0f; sFeat[wv][lane][10] = 0.0f; sFeat[wv][lane][11] = 0.0f;
    __syncthreads();

    const int half = lane >> 4;                      // K-pair selector per ISA layout
    const int l16  = lane & 15;
    const int cb   = coors[p * 3 + 0];
    const int cxi  = coors[p * 3 + 1];
    const int cyi  = coors[p * 3 + 2];

    #pragma unroll
    for (int nt = 0; nt < 4; nt++) {                 // 4 x 16 output channels
        v8f c0 = {};                                 // points 0..15
        v8f c1 = {};                                 // points 16..31
        #pragma unroll
        for (int ks = 0; ks < 3; ks++) {             // K = 12 in steps of 4
            const int k0 = ks * 4 + half * 2;
            v2f a0, a1, b;
            a0[0] = sFeat[wv][l16][k0];        a0[1] = sFeat[wv][l16][k0 + 1];
            a1[0] = sFeat[wv][16 + l16][k0];   a1[1] = sFeat[wv][16 + l16][k0 + 1];
            b[0]  = sW[k0][nt * 16 + l16];     b[1]  = sW[k0 + 1][nt * 16 + l16];
            c0 = __builtin_amdgcn_wmma_f32_16x16x4_f32(false, a0, false, b,
                                                       (short)0, c0, false, false);
            c1 = __builtin_amdgcn_wmma_f32_16x16x4_f32(false, a1, false, b,
                                                       (short)0, c1, false, false);
        }
        // BN (folded) + ReLU per element, then max over the 32 points.
        const float sc = sScale[nt * 16 + l16];
        const float sh = sShift[nt * 16 + l16];
        float m = 0.0f;                              // ReLU outputs are >= 0
        #pragma unroll
        for (int r = 0; r < 8; r++) {
            const float v0 = fmaxf(c0[r] * sc + sh, 0.0f);
            const float v1 = fmaxf(c1[r] * sc + sh, 0.0f);
            m = fmaxf(m, fmaxf(v0, v1));
        }
        m = fmaxf(m, __shfl_xor(m, 16, 32));         // merge rows r and r+8
        if (half == 0) {                             // lanes 0..15 hold channels
            const int ch = nt * 16 + l16;
            const size_t o = (((size_t)cb * OUT_CH + ch) * Y_L + cyi) * X_L + cxi;
            __builtin_nontemporal_store(m, &out[o]); // scattered, don't pollute L2
        }
    }
}

extern "C" void kernel_launch(void* const* d_in, const int* in_sizes, int n_in,
                              void* d_out, int out_size, void* d_ws, size_t ws_size,
                              hipStream_t stream) {
    const float* pillars = (const float*)d_in[0];
    const int*   coors   = (const int*)d_in[1];
    const int*   npoints = (const int*)d_in[2];
    const float* W       = (const float*)d_in[3];
    const float* gamma   = (const float*)d_in[4];
    const float* beta    = (const float*)d_in[5];
    float* out = (float*)d_out;

    float* partials = (float*)d_ws;                          // 250*54 floats
    float* scale    = partials + STATS_BLOCKS * NSTAT;       // 64
    float* shift    = scale + OUT_CH;                        // 64

    hipMemsetAsync(d_out, 0, (size_t)out_size * sizeof(float), stream);
    pillar_stats   <<<STATS_BLOCKS, 256, 0, stream>>>(pillars, coors, npoints, partials);
    pillar_finalize<<<1, 64, 0, stream>>>(partials, W, gamma, beta, scale, shift);
    pillar_main    <<<P_PILLARS / 8, 256, 0, stream>>>(pillars, coors, npoints, W,
                                                       scale, shift, out);
}